// TFSTP_49512382988539
// MI455X (gfx1250) — compile-verified
//
#include <hip/hip_runtime.h>
#include <stdint.h>

// Problem constants (reference: T=400, H=250, W=400, half=200)
#define T_TOT  400
#define HALF   200
#define HW     100000
#define TC     25                  // time steps per TDM tile
#define NCHUNK (T_TOT / TC)        // 16
#define BLK    256                 // pixels per block (1 pixel / thread)
#define NW     13                  // 13*32 = 416 bits >= 400

typedef unsigned int u32x4 __attribute__((ext_vector_type(4)));
typedef unsigned int u32x8 __attribute__((ext_vector_type(8)));

typedef __attribute__((address_space(3))) const char lds_cchar;

// Byte offset of a __shared__ object within the workgroup's LDS allocation
// (generic -> LDS addrspace cast, then ptrtoint => 32-bit LDS offset).
__device__ __forceinline__ unsigned lds_byte_off(const void* p) {
    return (unsigned)(unsigned long long)(lds_cchar*)p;
}

// Issue a TDM 2-D tile load: tile_h rows of tile_w f32 elements, row stride
// HW elements, from global 'gsrc' into LDS at byte offset 'ldsoff'.
// Descriptor layout per CDNA5 ISA §8 (groups 0/1; groups 2/3 NULL => 2-D).
__device__ __forceinline__ void tdm_load_tile(const float* gsrc, unsigned ldsoff,
                                              unsigned tile_w, unsigned tile_h) {
    unsigned long long ga = (unsigned long long)gsrc;
    const unsigned TD0 = 0x40000000u;   // huge tensor dims: never trips OOB clamp
    const unsigned TD1 = 0x40000000u;

    u32x4 g0;
    g0[0] = 1u;                                         // count=1, user descriptor
    g0[1] = ldsoff;                                     // lds_addr (bytes)
    g0[2] = (unsigned)ga;                               // global_addr[31:0]
    g0[3] = (unsigned)((ga >> 32) & 0x1FFFFFFull)       // global_addr[56:32]
          | (2u << 30);                                 // type = 2 ("image")

    u32x8 g1;
    g1[0] = (2u << 16);                                 // wg_mask=0, data_size=4B
    g1[1] = (TD0 & 0xFFFFu) << 16;                      // abar=0 | tensor_dim0 lo16
    g1[2] = (TD0 >> 16) | ((TD1 & 0xFFFFu) << 16);      // dim0 hi | dim1 lo
    g1[3] = (TD1 >> 16) | (tile_w << 16);               // dim1 hi | tile_dim0
    g1[4] = tile_h;                                     // tile_dim1 | tile_dim2=0
    g1[5] = (unsigned)HW;                               // tensor_dim0_stride lo32
    g1[6] = 0u;                                         // stride0 hi | stride1 lo
    g1[7] = 0u;                                         // stride1 hi
    asm volatile("tensor_load_to_lds %0, %1" :: "s"(g0), "s"(g1) : "memory");
}

// Order-preserving float <-> uint key (for atomic min/max of signed floats).
__device__ __forceinline__ unsigned fkey(float f) {
    unsigned u = __float_as_uint(f);
    return (u & 0x80000000u) ? ~u : (u | 0x80000000u);
}
__device__ __forceinline__ float unfkey(unsigned k) {
    return __uint_as_float((k & 0x80000000u) ? (k & 0x7FFFFFFFu) : ~k);
}

__global__ void tfstp_init_mm(unsigned* mm) {
    mm[0] = 0xFFFFFFFFu;   // min key
    mm[1] = 0u;            // max key
}

__global__ __launch_bounds__(BLK)
void tfstp_main(const float* __restrict__ spikes,
                float* __restrict__ out,
                unsigned* __restrict__ mm) {
    __shared__ float    stage[2][TC * BLK];       // double-buffered TDM tiles
    __shared__ unsigned mask[BLK * NW];           // per-pixel 416-bit spike mask

    const int tid = threadIdx.x;
    const int p0  = blockIdx.x * BLK;
    const int rem = HW - p0;
    const int wdt = (rem < BLK) ? rem : BLK;      // always a multiple of 32
    const bool active = tid < wdt;

    // ---------------- Phase 1: TDM-staged load + bit packing ----------------
    if (tid == 0)
        tdm_load_tile(spikes + p0, lds_byte_off(&stage[0][0]), (unsigned)wdt, TC);

    unsigned accw = 0u;
    int accwi = 0;

    for (int k = 0; k < NCHUNK; ++k) {
        if (tid == 0)
            __builtin_amdgcn_s_wait_tensorcnt(0); // chunk k is resident in LDS
        __syncthreads();                          // publish to all waves; prev buf free
        if (tid == 0 && (k + 1) < NCHUNK)
            tdm_load_tile(spikes + (size_t)(k + 1) * TC * HW + p0,
                          lds_byte_off(&stage[(k + 1) & 1][0]),
                          (unsigned)wdt, TC);     // overlap next DMA with packing
        if (active) {
            const float* buf = &stage[k & 1][0];
#pragma unroll
            for (int tt = 0; tt < TC; ++tt) {
                const int t  = k * TC + tt;
                const int wi = t >> 5;
                if (wi != accwi) {                // t is uniform -> scalar control
                    mask[tid * NW + accwi] = accw;
                    accw = 0u;
                    accwi = wi;
                }
                if (buf[tt * wdt + tid] > 0.0f)
                    accw |= 1u << (t & 31);
            }
        }
    }

    if (!active) return;
    mask[tid * NW + accwi] = accw;                // flush word 12 (bits 400.. = 0)

    // ---------------- Phase 2: ISI scan, 200-step (R,u) recurrence ----------
    const int mb = tid * NW;
    int nxt = -1;                                 // first spike index >= t
    int prev = -1;                                // last spike index  <  t
    unsigned curw = mask[mb];
    unsigned spk_prev = curw & 1u;
    float R = 1.0f, u = 0.15f;

    for (int t = 1; t <= HALF; ++t) {
        if ((t & 31) == 0) curw = mask[mb + (t >> 5)];
        if (spk_prev) prev = t - 1;
        if (nxt < t) {                            // advance monotone next-spike ptr
            int wi = t >> 5;
            unsigned b = mask[mb + wi] & (0xFFFFFFFFu << (t & 31));
            while (b == 0u && ++wi < NW) b = mask[mb + wi];
            nxt = b ? ((wi << 5) + __builtin_ctz(b)) : 1024;
        }
        const unsigned spk = (curw >> (t & 31)) & 1u;
        const int  di  = nxt - prev;
        // isi[0] is always inf, so (isi != isi0) == isfinite(isi);
        // mask = (finite & !spk) | (isi == 1)  ==  finite & (!spk | di==1)
        const bool upd = (prev >= 0) && (nxt < T_TOT) && (!spk || di == 1);
        const float isi = (float)di;
        const float eD = __expf(-isi);            // D = 1.0
        const float eF = __expf(-isi * 0.1f);     // F = 10.0
        const float Rn = 1.0f - (1.0f - R * (1.0f - u)) * eD;
        const float un = 0.15f + (0.85f * u) * eF; // U0 + (u+f(1-u)-U0) eF, f=U0=0.15
        R = upd ? Rn : R;
        u = upd ? un : u;
        spk_prev = spk;
    }

    // ---------------- Phase 3: rho image + global min/max -------------------
    const float rho_u = -1.0f / (10.0f * __logf((u - 0.15f) / (9.85f + 0.85f * u)));
    const float rho_R = -1.0f / (__logf((1.0f - R) / (1.0f - R * (1.0f - u))));
    const float img = rho_u + rho_R;
    out[p0 + tid] = img;

    float vmn = img, vmx = img;                   // active waves are always full
#pragma unroll
    for (int off = 16; off > 0; off >>= 1) {
        vmn = fminf(vmn, __shfl_xor(vmn, off, 32));
        vmx = fmaxf(vmx, __shfl_xor(vmx, off, 32));
    }
    if ((tid & 31) == 0) {
        atomicMin(&mm[0], fkey(vmn));
        atomicMax(&mm[1], fkey(vmx));
    }
}

__global__ __launch_bounds__(BLK)
void tfstp_normalize(float* __restrict__ out, const unsigned* __restrict__ mm) {
    const int i = blockIdx.x * BLK + threadIdx.x;
    if (i >= HW) return;
    const float mn = unfkey(mm[0]);
    const float mx = unfkey(mm[1]);
    const float v = out[i];
    out[i] = (mx != mn) ? (v - mn) / (mx - mn) : v;
}

extern "C" void kernel_launch(void* const* d_in, const int* in_sizes, int n_in,
                              void* d_out, int out_size, void* d_ws, size_t ws_size,
                              hipStream_t stream) {
    (void)in_sizes; (void)n_in; (void)out_size; (void)ws_size;
    const float* spikes = (const float*)d_in[0];   // [1,400,250,400] f32
    float* out = (float*)d_out;                    // [1,1,250,400]  f32
    unsigned* mm = (unsigned*)d_ws;                // 2 x u32 min/max keys

    const int nblk = (HW + BLK - 1) / BLK;         // 391
    tfstp_init_mm<<<1, 1, 0, stream>>>(mm);
    tfstp_main<<<nblk, BLK, 0, stream>>>(spikes, out, mm);
    tfstp_normalize<<<nblk, BLK, 0, stream>>>(out, mm);
}